// StructFun_65816078844198
// MI455X (gfx1250) — compile-verified
//
#include <hip/hip_runtime.h>

// Fused FractionalMaxPool(3x3,s2,p1) + depthwise 7x7 correlation (CDNA5 WMMA).
//
// Roofline: >=129MB HBM traffic (read x 103MB, write out 26MB) -> ~5.5us floor
// at 23.3TB/s; FLOPs ~0.7G (fp32) -> memory bound. One fused kernel: pool
// directly into LDS (pooled tensor never touches HBM), then a banded-matrix
// implicit GEMM runs the 7x7 depthwise correlation on fp32-exact
// V_WMMA_F32_16X16X4_F32:  O(16x16) = sum_i A_i(16x22) * B_i(22x16).
//
// v3 changes vs v2:
//  - Pooling stays branch-free (clamp == -inf pad for max), but exploits that
//    x1=2C / x2=2C+1 and y1=2R / y2=2R+1 are always in-bounds: each window row
//    is one aligned v2f load (global_load_b64) + one clamped b32 left tap.
//    6 unconditional, clause-able loads per pooled element (was 9 scalar),
//    with much less clamp/address VALU.

typedef __attribute__((ext_vector_type(2))) float v2f;
typedef __attribute__((ext_vector_type(8))) float v8f;

#define NB 16      // batch
#define ND 32      // channels
#define HX 224     // x height
#define WX 224     // x width
#define HS 112     // pooled/out height
#define WS 112     // pooled/out width
#define FH 7
#define FW 7
#define TILE 16
#define TROWS (HS / TILE)     // 7 tile-rows per plane
#define NWAVE (WS / TILE)     // 7 waves, one 16x16 tile each
#define BLOCK (NWAVE * 32)    // 224 threads
#define PROWS 22              // 16 output rows + 6 halo rows
#define PPITCH 122            // >=120 needed; 122 dwords -> ~conflict-free
#define KPAD 24               // K = 22 padded to multiple of 4

__global__ __launch_bounds__(BLOCK) void fused_pool_dwconv_wmma(
    const float* __restrict__ x, const float* __restrict__ S,
    float* __restrict__ out)
{
    __shared__ float Pool[PROWS * PPITCH];   // pooled strip incl. zero halo
    __shared__ float Bm[FH * KPAD * 16];     // banded kernel matrices

    const int b   = blockIdx.x;
    const int tr  = b % TROWS;
    const int d   = (b / TROWS) % ND;
    const int n   = b / (TROWS * ND);
    const int r0  = tr * TILE;
    const int tid = threadIdx.x;

    const float* xp = x + ((size_t)(n * ND + d)) * (HX * WX);
    const float* sp = S + d * (FH * FW);

    // ---- Stage 1: 3x3 stride-2 pad-1 max-pool of x -> LDS strip (branch-free).
    // Pool[pr][cc] = pooled[r0-3+pr, cc-3], 0.0 outside (conv zero-halo).
    // Clamped duplicate taps are a no-op under max (== -inf padding). For a
    // valid (Rc,Cc): rows 2Rc and 2Rc+1 and cols 2Cc, 2Cc+1 are always
    // in-bounds, so each window row = 1 aligned v2f load + 1 clamped left tap.
    for (int e = tid; e < PROWS * PPITCH; e += BLOCK) {
        const int pr = e / PPITCH;
        const int cc = e % PPITCH;
        const int R  = r0 - 3 + pr;                 // pooled row (may be OOB)
        const int C  = cc - 3;                      // pooled col (may be OOB)
        const int Rc = R < 0 ? 0 : (R >= HS ? HS - 1 : R);
        const int Cc = C < 0 ? 0 : (C >= WS ? WS - 1 : C);
        const int ym = 2 * Rc - 1 < 0 ? 0 : 2 * Rc - 1;   // clamped top row
        const int xm = 2 * Cc - 1 < 0 ? 0 : 2 * Cc - 1;   // clamped left col
        const float* rowm = xp + ym * WX;                  // y = 2Rc-1 (clamped)
        const float* row0 = xp + 2 * Rc * WX;              // y = 2Rc
        const float* rowp = row0 + WX;                     // y = 2Rc+1 (<=223)
        const v2f pm = *(const v2f*)(rowm + 2 * Cc);       // cols 2Cc, 2Cc+1
        const v2f p0 = *(const v2f*)(row0 + 2 * Cc);
        const v2f pp = *(const v2f*)(rowp + 2 * Cc);
        const float lm = rowm[xm];
        const float l0 = row0[xm];
        const float lp = rowp[xm];
        float m = fmaxf(fmaxf(pm.x, pm.y), lm);
        m = fmaxf(m, fmaxf(fmaxf(p0.x, p0.y), l0));
        m = fmaxf(m, fmaxf(fmaxf(pp.x, pp.y), lp));
        const bool valid = (R >= 0) & (R < HS) & (C >= 0) & (C < WS);
        Pool[pr * PPITCH + cc] = valid ? m : 0.0f;  // halo must be exactly 0
    }

    // ---- Stage 2: banded B matrices: Bm[i][k][c] = S[i][k-c] in-band else 0.
    // Explicit zeros everywhere (incl. K pad rows 22..23) so no NaN*0 hazard.
    for (int e = tid; e < FH * KPAD * 16; e += BLOCK) {
        const int i = e / (KPAD * 16);
        const int k = (e / 16) % KPAD;
        const int c = e % 16;
        const int j = k - c;
        float v = 0.0f;
        if (j >= 0 && j < FW && k < 22) v = sp[i * FW + j];
        Bm[e] = v;
    }
    __syncthreads();

    // ---- Stage 3: one 16x16 output tile per wave, 42x V_WMMA_F32_16X16X4_F32.
    // A frag (16x4 f32): lane = M%16 + 16*(K>=2), vgpr = K%2.
    // B frag (4x16):     lane = N%16 + 16*(K>=2), vgpr = K%2.
    // C/D (16x16 f32):   vgpr v -> M = v + 8*(lane>=16), N = lane%16.
    const int wave  = tid >> 5;        // 0..6 -> tile column
    const int lane  = tid & 31;
    const int mrow  = lane & 15;       // A row (M), also B col (N)
    const int khalf = lane >> 4;       // K pair {0,1} vs {2,3}
    const int cbase = wave * TILE;

    v8f acc = {0.f, 0.f, 0.f, 0.f, 0.f, 0.f, 0.f, 0.f};
#pragma unroll
    for (int i = 0; i < FH; ++i) {
        // base index is even (PPITCH, cbase, khalf*2, kb*4 all even) -> the
        // v2f load below is 8B-aligned and lowers to ds_load_b64.
        const float* arow = &Pool[(mrow + i) * PPITCH + cbase + khalf * 2];
        const float* bcol = &Bm[i * (KPAD * 16) + khalf * 2 * 16 + mrow];
#pragma unroll
        for (int kb = 0; kb < KPAD / 4; ++kb) {
            v2f a = *(const v2f*)(arow + kb * 4);
            v2f bv;
            bv.x = bcol[kb * 4 * 16];
            bv.y = bcol[kb * 4 * 16 + 16];
            acc = __builtin_amdgcn_wmma_f32_16x16x4_f32(
                false, a, false, bv, (short)0, acc, false, false);
        }
    }

    // ---- Store: lanes 0..15 write row M=v, lanes 16..31 row M=v+8 (64B rows).
    float* op = out + ((size_t)(n * ND + d)) * (HS * WS)
                    + (size_t)r0 * WS + cbase;
#pragma unroll
    for (int v = 0; v < 8; ++v) {
        const int m = v + 8 * khalf;
        op[m * WS + mrow] = acc[v];
    }
}

extern "C" void kernel_launch(void* const* d_in, const int* in_sizes, int n_in,
                              void* d_out, int out_size, void* d_ws, size_t ws_size,
                              hipStream_t stream) {
    (void)in_sizes; (void)n_in; (void)d_ws; (void)ws_size; (void)out_size;
    const float* x = (const float*)d_in[0];   // (16,32,224,224) f32
    const float* S = (const float*)d_in[1];   // (1,32,7,7) f32
    float* out = (float*)d_out;               // (16,32,1,112*112) f32

    dim3 grid(NB * ND * TROWS);               // 3584 blocks
    fused_pool_dwconv_wmma<<<grid, BLOCK, 0, stream>>>(x, S, out);
}